// EntropyBottleneck_13288628814287
// MI455X (gfx1250) — compile-verified
//
#include <hip/hip_runtime.h>
#include <stdint.h>
#include <math.h>

// ---------------------------------------------------------------------------
// EntropyBottleneck forward for MI455X (gfx1250).
//   x, noise : (1,128,48,48,48) f32
//   out      : [outputs (14155776) | likelihood (14155776) | qloss (1)]
// Roofline: 226 MB traffic => ~9.7us @ 23.3 TB/s; ~310M transcendentals
// (18 tanh + 2 exp + 2 rcp per element) => transcendental-bound at the
// margin. gfx1250 v_tanh_f32 / v_exp_f32 / v_rcp_f32 give 1-instr trans ops;
// inputs staged via CDNA5 async global->LDS copies (ASYNCcnt pipeline).
// ---------------------------------------------------------------------------

typedef float v4f __attribute__((ext_vector_type(4)));

#define C_CH 128
#define SPATIAL 110592                  // 48*48*48
#define TOTAL_ELEMS (C_CH * SPATIAL)    // 14155776
#define BLK_ELEMS 4096
#define BLOCKS_PER_CH (SPATIAL / BLK_ELEMS)   // 27
#define NBLOCKS (C_CH * BLOCKS_PER_CH)        // 3456
#define PSTRIDE 48

// Per-channel params, layer-0 bias pre-shifted by -/+ 0.5*w0 (46 floats).
// Layout in d_ws (stride 48):
//  0..2  w0      3..5  bL (b0-0.5w0)   6..8  bU (b0+0.5w0)   9..11 t0
// 12..20 W1     21..23 b1             24..26 t1
// 27..35 W2     36..38 b2             39..41 t2
// 42..44 w3     45     b3
struct EP {
    float w0[3], bL[3], bU[3], t0[3];
    float W1[9], bb1[3], t1[3];
    float W2[9], bb2[3], t2[3];
    float w3[3], bb3;
};

__device__ __forceinline__ float fast_tanh(float x) {
    float r;
    asm("v_tanh_f32 %0, %1" : "=v"(r) : "v"(x));   // gfx1250 VOP1 transcendental
    return r;
}

__device__ __forceinline__ float fast_sig(float x) {
    // sigmoid(x) = 1 / (1 + exp(-x)); exp(-x) = exp2(-x*log2(e))
    float e = __builtin_amdgcn_exp2f(-1.44269504088896340736f * x);
    return __builtin_amdgcn_rcpf(1.0f + e);
}

__device__ __forceinline__ float bcast(float v) {
    // force block-uniform params into SGPRs
    return __uint_as_float(__builtin_amdgcn_readfirstlane(__float_as_uint(v)));
}

// Numerically stable softplus on HW exp2/log2:
//   softplus(x) = max(x,0) + ln2 * log2(1 + exp2(-|x|*log2e))
__device__ __forceinline__ float softplus_f(float x) {
    float t = __builtin_amdgcn_exp2f(-fabsf(x) * 1.44269504088896340736f);
    float l = __builtin_amdgcn_logf(1.0f + t) * 0.69314718055994530942f;
    return fmaxf(x, 0.0f) + l;
}

// 1 -> 3 -> 3 -> 3 -> 1 per-channel CDF MLP, fully unrolled in registers.
// First-layer bias passed explicitly so callers can use raw/shifted variants.
__device__ __forceinline__ float eb_mlp_core(
    const float w0[3], const float b0[3], const float t0[3],
    const float W1[9], const float b1[3], const float t1[3],
    const float W2[9], const float b2[3], const float t2[3],
    const float w3[3], float b3, float z) {
    float v0 = fmaf(w0[0], z, b0[0]);
    float v1 = fmaf(w0[1], z, b0[1]);
    float v2 = fmaf(w0[2], z, b0[2]);
    float h0 = fmaf(t0[0], fast_tanh(v0), v0);
    float h1 = fmaf(t0[1], fast_tanh(v1), v1);
    float h2 = fmaf(t0[2], fast_tanh(v2), v2);

    float a0 = fmaf(W1[0], h0, fmaf(W1[1], h1, fmaf(W1[2], h2, b1[0])));
    float a1 = fmaf(W1[3], h0, fmaf(W1[4], h1, fmaf(W1[5], h2, b1[1])));
    float a2 = fmaf(W1[6], h0, fmaf(W1[7], h1, fmaf(W1[8], h2, b1[2])));
    float g0 = fmaf(t1[0], fast_tanh(a0), a0);
    float g1 = fmaf(t1[1], fast_tanh(a1), a1);
    float g2 = fmaf(t1[2], fast_tanh(a2), a2);

    float y0 = fmaf(W2[0], g0, fmaf(W2[1], g1, fmaf(W2[2], g2, b2[0])));
    float y1 = fmaf(W2[3], g0, fmaf(W2[4], g1, fmaf(W2[5], g2, b2[1])));
    float y2 = fmaf(W2[6], g0, fmaf(W2[7], g1, fmaf(W2[8], g2, b2[2])));
    float u0 = fmaf(t2[0], fast_tanh(y0), y0);
    float u1 = fmaf(t2[1], fast_tanh(y1), y1);
    float u2 = fmaf(t2[2], fast_tanh(y2), y2);

    return fmaf(w3[0], u0, fmaf(w3[1], u1, fmaf(w3[2], u2, b3)));
}

__device__ __forceinline__ float eb_lik(const EP& e, float o) {
    // lower = mlp(o-0.5) via pre-shifted bias bL; upper via bU (both at z=o)
    float lo = eb_mlp_core(e.w0, e.bL, e.t0, e.W1, e.bb1, e.t1,
                           e.W2, e.bb2, e.t2, e.w3, e.bb3, o);
    float up = eb_mlp_core(e.w0, e.bU, e.t0, e.W1, e.bb1, e.t1,
                           e.W2, e.bb2, e.t2, e.w3, e.bb3, o);
    float sum = lo + up;
    float s = (sum > 0.0f) ? -1.0f : ((sum < 0.0f) ? 1.0f : 0.0f);
    float a = fast_sig(s * up);
    float b = fast_sig(s * lo);
    return fmaxf(fabsf(a - b), 1e-9f);
}

// ---------------------------------------------------------------------------
// Kernel 1: per-channel param precompute (softplus/tanh, bias fusion) +
//           quantiles loss with LDS tree reduce.
// ---------------------------------------------------------------------------
__global__ void __launch_bounds__(128) eb_setup(
    const float* __restrict__ m0, const float* __restrict__ b0, const float* __restrict__ f0,
    const float* __restrict__ m1, const float* __restrict__ b1, const float* __restrict__ f1,
    const float* __restrict__ m2, const float* __restrict__ b2, const float* __restrict__ f2,
    const float* __restrict__ m3, const float* __restrict__ b3, const float* __restrict__ q,
    float* __restrict__ P, float* __restrict__ qloss) {
    const int c = threadIdx.x;

    float w0[3], bb0[3], t0[3], W1[9], bb1[3], t1[3], W2[9], bb2[3], t2[3], w3[3];
#pragma unroll
    for (int j = 0; j < 3; ++j) {
        w0[j]  = softplus_f(m0[c * 3 + j]);
        bb0[j] = b0[c * 3 + j];
        t0[j]  = fast_tanh(f0[c * 3 + j]);
        bb1[j] = b1[c * 3 + j];
        t1[j]  = fast_tanh(f1[c * 3 + j]);
        bb2[j] = b2[c * 3 + j];
        t2[j]  = fast_tanh(f2[c * 3 + j]);
        w3[j]  = softplus_f(m3[c * 3 + j]);
    }
#pragma unroll
    for (int k = 0; k < 9; ++k) {
        W1[k] = softplus_f(m1[c * 9 + k]);
        W2[k] = softplus_f(m2[c * 9 + k]);
    }
    const float bb3 = b3[c];

    float* dst = P + c * PSTRIDE;
#pragma unroll
    for (int j = 0; j < 3; ++j) {
        dst[0 + j]  = w0[j];
        dst[3 + j]  = fmaf(-0.5f, w0[j], bb0[j]);   // bL
        dst[6 + j]  = fmaf( 0.5f, w0[j], bb0[j]);   // bU
        dst[9 + j]  = t0[j];
        dst[21 + j] = bb1[j];
        dst[24 + j] = t1[j];
        dst[36 + j] = bb2[j];
        dst[39 + j] = t2[j];
        dst[42 + j] = w3[j];
    }
#pragma unroll
    for (int k = 0; k < 9; ++k) {
        dst[12 + k] = W1[k];
        dst[27 + k] = W2[k];
    }
    dst[45] = bb3;

    // quantiles loss: sum |mlp(quantiles) - [-T, 0, T]| (raw bb0, no shift)
    const float T = logf(2.0f / 1e-9f - 1.0f);
    float acc =
        fabsf(eb_mlp_core(w0, bb0, t0, W1, bb1, t1, W2, bb2, t2, w3, bb3, q[c * 3 + 0]) + T) +
        fabsf(eb_mlp_core(w0, bb0, t0, W1, bb1, t1, W2, bb2, t2, w3, bb3, q[c * 3 + 1])) +
        fabsf(eb_mlp_core(w0, bb0, t0, W1, bb1, t1, W2, bb2, t2, w3, bb3, q[c * 3 + 2]) - T);

    __shared__ float red[128];
    red[c] = acc;
    __syncthreads();
#pragma unroll
    for (int s = 64; s > 0; s >>= 1) {
        if (c < s) red[c] += red[c + s];
        __syncthreads();
    }
    if (c == 0) qloss[0] = red[0];
}

// ---------------------------------------------------------------------------
// Kernel 2: streaming elementwise kernel; async global->LDS staged pipeline.
// Each thread issues and consumes its own B128 async copies (no barriers).
// ---------------------------------------------------------------------------
__device__ __forceinline__ void async_b128(uint32_t lds_addr, const float* gptr) {
    asm volatile("global_load_async_to_lds_b128 %0, %1, off"
                 :: "v"(lds_addr), "v"(gptr)
                 : "memory");
}

__global__ void __launch_bounds__(256) eb_main(
    const float* __restrict__ x, const float* __restrict__ noise,
    const float* __restrict__ P, float* __restrict__ out) {
    __shared__ __align__(16) float sx[BLK_ELEMS];
    __shared__ __align__(16) float sn[BLK_ELEMS];

    const int b = blockIdx.x;
    const int ch = b / BLOCKS_PER_CH;
    const int blk = b - ch * BLOCKS_PER_CH;
    const int t = threadIdx.x;
    const long base = (long)ch * SPATIAL + (long)blk * BLK_ELEMS;

    // Issue all 8 async copies up-front (4 chunks x {x, noise}); ASYNCcnt = 8.
    {
        const float* gx = x + base + t * 4;
        const float* gn = noise + base + t * 4;
        uint32_t lx = (uint32_t)(size_t)sx + (uint32_t)t * 16u;
        uint32_t ln = (uint32_t)(size_t)sn + (uint32_t)t * 16u;
#pragma unroll
        for (int k = 0; k < 4; ++k) {
            async_b128(lx + (uint32_t)k * 4096u, gx + k * 1024);
            async_b128(ln + (uint32_t)k * 4096u, gn + k * 1024);
        }
    }

    // Block-uniform per-channel params -> SGPRs while copies are in flight.
    const float* pp = P + ch * PSTRIDE;
    EP e;
#pragma unroll
    for (int j = 0; j < 3; ++j) {
        e.w0[j]  = bcast(pp[0 + j]);
        e.bL[j]  = bcast(pp[3 + j]);
        e.bU[j]  = bcast(pp[6 + j]);
        e.t0[j]  = bcast(pp[9 + j]);
        e.bb1[j] = bcast(pp[21 + j]);
        e.t1[j]  = bcast(pp[24 + j]);
        e.bb2[j] = bcast(pp[36 + j]);
        e.t2[j]  = bcast(pp[39 + j]);
        e.w3[j]  = bcast(pp[42 + j]);
    }
#pragma unroll
    for (int k = 0; k < 9; ++k) {
        e.W1[k] = bcast(pp[12 + k]);
        e.W2[k] = bcast(pp[27 + k]);
    }
    e.bb3 = bcast(pp[45]);

    // Drain the pipeline: chunk k ready when asynccnt <= 8-2*(k+1).
#pragma unroll
    for (int k = 0; k < 4; ++k) {
        if (k == 0)      asm volatile("s_wait_asynccnt 6" ::: "memory");
        else if (k == 1) asm volatile("s_wait_asynccnt 4" ::: "memory");
        else if (k == 2) asm volatile("s_wait_asynccnt 2" ::: "memory");
        else             asm volatile("s_wait_asynccnt 0" ::: "memory");

        v4f xv = *(const v4f*)(sx + (k * 1024 + t * 4));
        v4f nv = *(const v4f*)(sn + (k * 1024 + t * 4));

        v4f ov, lv;
#pragma unroll
        for (int j = 0; j < 4; ++j) {
            float o = xv[j] + nv[j] - 0.5f;
            ov[j] = o;
            lv[j] = eb_lik(e, o);
        }

        const long idx = base + (long)(k * 1024 + t * 4);
        __builtin_nontemporal_store(ov, (v4f*)(out + idx));                       // outputs
        __builtin_nontemporal_store(lv, (v4f*)(out + (long)TOTAL_ELEMS + idx));   // likelihood
    }
}

// ---------------------------------------------------------------------------
extern "C" void kernel_launch(void* const* d_in, const int* in_sizes, int n_in,
                              void* d_out, int out_size, void* d_ws, size_t ws_size,
                              hipStream_t stream) {
    const float* x  = (const float*)d_in[0];
    const float* nz = (const float*)d_in[1];
    const float* m0 = (const float*)d_in[2];
    const float* b0 = (const float*)d_in[3];
    const float* f0 = (const float*)d_in[4];
    const float* m1 = (const float*)d_in[5];
    const float* b1 = (const float*)d_in[6];
    const float* f1 = (const float*)d_in[7];
    const float* m2 = (const float*)d_in[8];
    const float* b2 = (const float*)d_in[9];
    const float* f2 = (const float*)d_in[10];
    const float* m3 = (const float*)d_in[11];
    const float* b3 = (const float*)d_in[12];
    const float* q  = (const float*)d_in[13];

    float* out = (float*)d_out;
    float* P   = (float*)d_ws;  // 128 * 48 * 4 = 24 KB of scratch

    eb_setup<<<1, 128, 0, stream>>>(m0, b0, f0, m1, b1, f1, m2, b2, f2, m3, b3, q,
                                    P, out + 2L * TOTAL_ELEMS);
    eb_main<<<NBLOCKS, 256, 0, stream>>>(x, nz, P, out);
}